// PwlNetwork_23527830848188
// MI455X (gfx1250) — compile-verified
//
#include <hip/hip_runtime.h>

typedef __attribute__((ext_vector_type(2))) float v2f;
typedef __attribute__((ext_vector_type(8))) float v8f;
typedef int v4i __attribute__((vector_size(16)));   // matches builtin's param type

#define NUM_BIN        64
#define NUM_CAT_GROUPS 64
#define INTERVALS      16
#define NUM_CAT_CH     (NUM_CAT_GROUPS * INTERVALS)          // 1024
#define NUM_NUM        4096
#define KVAL           16
#define BATCH          16384
#define IN1            (NUM_BIN + NUM_CAT_CH + NUM_NUM)      // 5184
#define IN2            (NUM_BIN + NUM_CAT_GROUPS + NUM_NUM / KVAL) // 384

#define KC       64                  // K-chunk per stage (5184 = 81 * 64)
#define NCHUNK   (IN1 / KC)          // 81
#define TSTRIDE  68                  // padded LDS row stride (floats) -> conflict-free A reads
#define ROWS     16
#define WAVES    2                   // waves per block (share the weight vector in LDS)

// ---------------- CDNA5 async global->LDS path (guarded) ----------------
#if __has_builtin(__builtin_amdgcn_global_load_async_to_lds_b128) && \
    __has_builtin(__builtin_amdgcn_global_load_async_to_lds_b32)
#define HAS_ASYNC 1
#else
#define HAS_ASYNC 0
#endif

__device__ __forceinline__ void async_cp16(const float* g, float* l) {
#if HAS_ASYNC
    __builtin_amdgcn_global_load_async_to_lds_b128(
        (__attribute__((address_space(1))) v4i*)(uintptr_t)g,
        (__attribute__((address_space(3))) v4i*)(uint32_t)(uintptr_t)l,
        0, 0);
#else
    *(float4*)l = *(const float4*)g;   // global_load_b128 + ds_store_b128
#endif
}

__device__ __forceinline__ void async_cp4(const float* g, float* l) {
#if HAS_ASYNC
    __builtin_amdgcn_global_load_async_to_lds_b32(
        (__attribute__((address_space(1))) int*)(uintptr_t)g,
        (__attribute__((address_space(3))) int*)(uint32_t)(uintptr_t)l,
        0, 0);
#else
    *l = *g;
#endif
}

template <int N>
__device__ __forceinline__ void wait_async() {
#if HAS_ASYNC
#if __has_builtin(__builtin_amdgcn_s_wait_asynccnt)
    __builtin_amdgcn_s_wait_asynccnt(N);
#else
    asm volatile("s_wait_asynccnt %0" :: "i"(N) : "memory");
#endif
#endif
    asm volatile("" ::: "memory");     // compiler ordering fence
}

// ---------------- prep: fold network into wEff[] + scalar C ----------------
__device__ __forceinline__ int agg_index(int i) {
    if (i < NUM_BIN) return i;
    if (i < NUM_BIN + NUM_CAT_CH) return NUM_BIN + ((i - NUM_BIN) >> 4);
    return NUM_BIN + NUM_CAT_GROUPS + ((i - NUM_BIN - NUM_CAT_CH) >> 4);
}

__global__ void pwl_prep(const float* __restrict__ w1, const float* __restrict__ b1,
                         const float* __restrict__ w2, const float* __restrict__ b2,
                         float* __restrict__ wEff, float* __restrict__ cOut) {
    __shared__ float red[256];
    const int t = threadIdx.x;
    float acc = 0.f;
    for (int i = t; i < IN1; i += 256) {
        const int m  = agg_index(i);
        const float w2m = w2[m];
        wEff[i] = w1[i] * w2m;
        acc    += b1[i] * w2m;
    }
    for (int j = t; j < IN2; j += 256) acc += b2[j];
    red[t] = acc;
    __syncthreads();
    for (int s = 128; s > 0; s >>= 1) {
        if (t < s) red[t] += red[t + s];
        __syncthreads();
    }
    if (t == 0) cOut[0] = red[0];
}

// ---------------- main GEMV: out[b] = dot(x[b,:], wEff) + C ----------------
__device__ __forceinline__ void stage_chunk(const float* __restrict__ x, int rowbase,
                                            int kbase, float* tbuf, int lane) {
#if HAS_ASYNC
    // WAR: make sure prior DS reads of this buffer have completed before the
    // async engine (tracked on a different counter) can overwrite it.
    asm volatile("s_wait_dscnt 0x0" ::: "memory");
#endif
    const int half = lane >> 4;          // lanes 0-15 -> even row, 16-31 -> odd row
    const int q    = lane & 15;          // 16-byte column slot
    #pragma unroll
    for (int r = 0; r < ROWS; r += 2) {
        const int row = r + half;
        async_cp16(x + (size_t)(rowbase + row) * IN1 + kbase + q * 4,
                   tbuf + row * TSTRIDE + q * 4);
    }
}

__device__ __forceinline__ void compute_chunk(const float* tbuf, const float* wch,
                                              int m, int khalf, v8f& acc) {
    #pragma unroll
    for (int kk = 0; kk < KC; kk += 4) {
        v2f a = *(const v2f*)(tbuf + m * TSTRIDE + kk + khalf);
        v2f b = *(const v2f*)(wch + kk + khalf);
        acc = __builtin_amdgcn_wmma_f32_16x16x4_f32(
            false, a, false, b, (short)0, acc, false, false);
    }
}

__global__ void __launch_bounds__(32 * WAVES)
pwl_gemv(const float* __restrict__ x, const float* __restrict__ wEff,
         const float* __restrict__ cPtr, float* __restrict__ out) {
    __shared__ __align__(16) float s_w[IN1];                            // 20736 B
    __shared__ __align__(16) float s_tiles[WAVES * 2 * ROWS * TSTRIDE]; // 17408 B

    const int tid  = threadIdx.x;
    const int wave = tid >> 5;
    const int lane = tid & 31;
    const int rowbase = (blockIdx.x * WAVES + wave) * ROWS;
    const int m     = lane & 15;              // A-fragment row
    const int khalf = (lane >> 4) << 1;       // lanes 0-15: K0/K1, 16-31: K2/K3

    // Stage folded weight vector once per block (block-cooperative, 5184 floats).
    {
        const int nt = 32 * WAVES;            // 64 threads
        #pragma unroll
        for (int i = 0; i < 20; ++i)          // 20 * 256 = 5120 floats
            async_cp16(wEff + i * nt * 4 + tid * 4, s_w + i * nt * 4 + tid * 4);
        async_cp4(wEff + 5120 + tid, s_w + 5120 + tid);  // remaining 64
    }

    float* t0 = s_tiles + wave * (2 * ROWS * TSTRIDE);
    float* t1 = t0 + ROWS * TSTRIDE;

    v8f acc = {};
    stage_chunk(x, rowbase, 0, t0, lane);     // chunk 0 in flight with weights
    wait_async<ROWS / 2>();                   // <=8 outstanding => weights landed
    __syncthreads();                          // all waves' weight shares visible

    for (int c = 0; c < NCHUNK - 1; ++c) {
        float* cur = (c & 1) ? t1 : t0;
        float* nxt = (c & 1) ? t0 : t1;
        stage_chunk(x, rowbase, (c + 1) * KC, nxt, lane);
        wait_async<ROWS / 2>();               // chunk c complete (in-order)
        compute_chunk(cur, s_w + c * KC, m, khalf, acc);
    }
    wait_async<0>();
    compute_chunk(((NCHUNK - 1) & 1) ? t1 : t0, s_w + (NCHUNK - 1) * KC, m, khalf, acc);

    // D layout: VGPR r, lane 0 -> (M=r, N=0); lane 16 -> (M=r+8, N=0)
    const float cconst = cPtr[0];
    if (lane == 0) {
        #pragma unroll
        for (int r = 0; r < 8; ++r) out[rowbase + r] = acc[r] + cconst;
    } else if (lane == 16) {
        #pragma unroll
        for (int r = 0; r < 8; ++r) out[rowbase + 8 + r] = acc[r] + cconst;
    }
}

extern "C" void kernel_launch(void* const* d_in, const int* in_sizes, int n_in,
                              void* d_out, int out_size, void* d_ws, size_t ws_size,
                              hipStream_t stream) {
    const float* x  = (const float*)d_in[0];
    const float* w1 = (const float*)d_in[1];
    const float* b1 = (const float*)d_in[2];
    const float* w2 = (const float*)d_in[3];
    const float* b2 = (const float*)d_in[4];

    float* wEff = (float*)d_ws;        // IN1 floats
    float* cOut = wEff + IN1;          // 1 float
    float* out  = (float*)d_out;       // BATCH floats

    pwl_prep<<<1, 256, 0, stream>>>(w1, b1, w2, b2, wEff, cOut);
    pwl_gemv<<<BATCH / (ROWS * WAVES), 32 * WAVES, 0, stream>>>(x, wEff, cOut, out);
}